// Onehot2Embed_66949950210384
// MI455X (gfx1250) — compile-verified
//
#include <hip/hip_runtime.h>
#include <stdint.h>

// Problem constants (from reference)
#define N_CLASSES 50000
#define EMBED_DIM 128
#define N_TOKENS  2048

// Phase-1 tiling: 3125 blocks * 32 chunks * 256 threads * 1 uint4
//               = 25,600,000 uint4 = 50000*2048 floats exactly.
#define P1_THREADS 256
#define P1_CHUNKS  32
#define P1_BLOCKS  3125

// ---------------------------------------------------------------------------
// Kernel 1 (placed first so its asm shows in the disasm snippet):
// stream the one-hot matrix [N_CLASSES, N_TOKENS] (row-major,
// flat = c*N_TOKENS + n) through LDS via async copies, double-buffered.
// Each lane copies its own 16B slot and reads back only its own slot, so no
// workgroup barriers are required — only s_wait_asynccnt.
// Recover idx[n] = c for the nonzero entries (exactly one per column n).
// ---------------------------------------------------------------------------
__global__ __launch_bounds__(P1_THREADS) void extract_idx_kernel(const unsigned int* __restrict__ types_u32,
                                                                 int* __restrict__ idx) {
    __shared__ uint4 buf[2][P1_THREADS];

    const unsigned int tid = threadIdx.x;

    // Wave-relative LDS byte addresses of this lane's two slots.
    // (low 32 bits of the generic pointer to LDS == LDS byte offset)
    const unsigned int lds0 = (unsigned int)(uintptr_t)&buf[0][tid];
    const unsigned int lds1 = (unsigned int)(uintptr_t)&buf[1][tid];

    const unsigned long long base = (unsigned long long)(uintptr_t)types_u32;

    // uint4 index of this thread in chunk 0 of this block
    const unsigned int vecBase = blockIdx.x * (P1_CHUNKS * P1_THREADS) + tid;

    // Prime the pipeline: async-copy chunk 0 into buf[0].
    {
        unsigned int off0 = vecBase * 16u;
        asm volatile("global_load_async_to_lds_b128 %0, %1, %2 offset:0 th:TH_LOAD_NT"
                     :: "v"(lds0), "v"(off0), "s"(base) : "memory");
    }

    for (int k = 0; k < P1_CHUNKS; ++k) {
        if (k + 1 < P1_CHUNKS) {
            // Issue chunk k+1 into the other buffer, then wait until only it
            // is outstanding (async loads complete in order -> chunk k done).
            unsigned int offn = (vecBase + (unsigned int)(k + 1) * P1_THREADS) * 16u;
            unsigned int ldsn = (k & 1) ? lds0 : lds1;  // buf[(k+1)&1]
            asm volatile("global_load_async_to_lds_b128 %0, %1, %2 offset:0 th:TH_LOAD_NT"
                         :: "v"(ldsn), "v"(offn), "s"(base) : "memory");
            asm volatile("s_wait_asynccnt 1" ::: "memory");
        } else {
            asm volatile("s_wait_asynccnt 0" ::: "memory");
        }

        uint4 v = buf[k & 1][tid];  // ds_load_b128 of this lane's own slot

        // one-hot: 1.0f == 0x3F800000, 0.0f == 0x0 -> integer nonzero test
        if (v.x | v.y | v.z | v.w) {
            unsigned int elem = (vecBase + (unsigned int)k * P1_THREADS) * 4u;
            unsigned int c = elem >> 11;       // / N_TOKENS (2048)
            unsigned int n = elem & 2047u;     // % N_TOKENS
            if (v.x) idx[n + 0] = (int)c;
            if (v.y) idx[n + 1] = (int)c;
            if (v.z) idx[n + 2] = (int)c;
            if (v.w) idx[n + 3] = (int)c;
        }
    }
}

// ---------------------------------------------------------------------------
// Kernel 0: warm the lookup table (25.6 MB) into L2 with global_prefetch_b8.
// Prefetches don't increment LOADcnt, so this kernel retires immediately and
// the L2 fill overlaps the phase-1 stream (which uses NT loads and thus does
// not evict it).
// ---------------------------------------------------------------------------
__global__ __launch_bounds__(256) void prefetch_lut_kernel(const char* __restrict__ lut_bytes,
                                                           int nbytes) {
    int line = blockIdx.x * blockDim.x + threadIdx.x;
    int byte = line * 128;  // one 128B cacheline per lane
    if (byte < nbytes) {
        __builtin_prefetch(lut_bytes + byte, 0, 1);  // -> global_prefetch_b8
    }
}

// ---------------------------------------------------------------------------
// Kernel 2: out[n, d] = lookup[d, idx[n]]  (lookup is [EMBED_DIM, N_CLASSES]
// row-major). One block per token, one lane per embedding dim. The scattered
// lookup reads hit L2 thanks to the prefetch kernel.
// ---------------------------------------------------------------------------
__global__ __launch_bounds__(EMBED_DIM) void gather_rows_kernel(const float* __restrict__ lut,
                                                                const int* __restrict__ idx,
                                                                float* __restrict__ out) {
    const int n = blockIdx.x;
    const int d = threadIdx.x;
    const int c = idx[n];  // uniform per block
    out[n * EMBED_DIM + d] = lut[d * N_CLASSES + c];
}

// ---------------------------------------------------------------------------
extern "C" void kernel_launch(void* const* d_in, const int* in_sizes, int n_in,
                              void* d_out, int out_size, void* d_ws, size_t ws_size,
                              hipStream_t stream) {
    const float* types = (const float*)d_in[0];  // [N_CLASSES, N_TOKENS] f32
    const float* lut   = (const float*)d_in[1];  // [EMBED_DIM, N_CLASSES] f32
    float* out = (float*)d_out;                  // [N_TOKENS, EMBED_DIM] f32
    int* idx = (int*)d_ws;                       // 2048 * 4B scratch

    // 0) warm lookup table into L2 (kernel retires immediately; fill overlaps
    //    the phase-1 stream).
    const int lut_bytes = EMBED_DIM * N_CLASSES * 4;
    const int lines = (lut_bytes + 127) / 128;
    prefetch_lut_kernel<<<(lines + 255) / 256, 256, 0, stream>>>((const char*)lut, lut_bytes);

    // 1) stream one-hot matrix, recover indices.
    extract_idx_kernel<<<P1_BLOCKS, P1_THREADS, 0, stream>>>((const unsigned int*)types, idx);

    // 2) gather embedding rows.
    gather_rows_kernel<<<N_TOKENS, EMBED_DIM, 0, stream>>>(lut, idx, out);
}